// RoiPooling_15221364097270
// MI455X (gfx1250) — compile-verified
//
#include <hip/hip_runtime.h>

typedef float v2f __attribute__((ext_vector_type(2)));
typedef float v8f __attribute__((ext_vector_type(8)));

#define C_DIM       1024
#define HW          128
#define LDS_STRIDE  132   // row-major T/I buffer: +4 pad -> conflict-free column access
#define PAIR_STRIDE 288   // k-pair interleaved F buffer: mod 64 == 32 -> lane halves disjoint
#define N_REGIONS   1024  // NUM_ROIS(256) * PATCH_NUM(4)

__global__ __launch_bounds__(256) void roi_pool_integral_wmma(
    const float* __restrict__ fmap,   // [C, H, W] fp32
    const float* __restrict__ roi,    // [256, 4]  fp32 (xmin,ymin,xmax,ymax)
    float* __restrict__ out)          // [N, C] fp32 (N=1024)
{
  // Pair-interleaved F: 64 * 288 floats = 72 KB; later reused row-major (128*132 <= 64*288)
  __shared__ float smem[64 * PAIR_STRIDE];

  const int tid  = threadIdx.x;
  const int lane = tid & 31;
  const int wave = tid >> 5;
  const int c    = blockIdx.x;

  // ---- Stage 0: load F_c, storing K-pair interleaved: smem[(k>>1)*S + 2n + (k&1)] ----
  {
    const float* src = fmap + (size_t)c * HW * HW;
#pragma unroll
    for (int it = 0; it < 16; ++it) {
      int idx = it * 256 + tid;        // 0..4095
      int j   = idx >> 6;              // k-pair row 0..63
      int ct  = idx & 63;              // column pair 0..63
      float2 r0 = *(const float2*)&src[(2 * j    ) * HW + 2 * ct];
      float2 r1 = *(const float2*)&src[(2 * j + 1) * HW + 2 * ct];
      float4 v  = make_float4(r0.x, r1.x, r0.y, r1.y);
      *(float4*)&smem[j * PAIR_STRIDE + 4 * ct] = v;   // single ds_store_b128
    }
  }
  __syncthreads();

  const int ml = lane & 15;          // M/N index within fragment
  const int kh = (lane >> 4) * 2;    // K sub-offset (0 or 2)
  const int dM = (lane >> 4) << 3;   // D-layout row offset (0 or 8)

  // Load-balanced wave->tile pairing: wave handles tile rows/cols {p, 7-p} over 4 tiles.
  const int p     = wave >> 1;       // 0..3
  const int h     = wave & 1;        // 0..1
  const int jBase = h * 4;

  v8f acc[8];
  const v8f vzero = {0.f, 0.f, 0.f, 0.f, 0.f, 0.f, 0.f, 0.f};

  // ---------- Stage 1: T = L * F   (inclusive column prefix over rows) ------
#pragma unroll
  for (int t = 0; t < 8; ++t) acc[t] = vzero;

#pragma unroll
  for (int half = 0; half < 2; ++half) {
    const int iRow  = half ? (7 - p) : p;     // output tile row
    const int kbMax = 4 * (iRow + 1);         // triangular skip: L[m,k]=0 for k>m
    const int mg    = iRow * 16 + ml;         // global row (A's M)
    for (int kb = 0; kb < kbMax; ++kb) {
      const int k0 = kb * 4 + kh;             // even
      v2f a;                                   // A = L fragment, generated
      a.x = (k0     <= mg) ? 1.0f : 0.0f;
      a.y = (k0 + 1 <= mg) ? 1.0f : 0.0f;
      const int rowOff = (k0 >> 1) * PAIR_STRIDE;
#pragma unroll
      for (int jj = 0; jj < 4; ++jj) {
        const int n = (jBase + jj) * 16 + ml;  // global col (B's N)
        v2f b = *(const v2f*)&smem[rowOff + 2 * n];   // one ds_load_b64, WMMA-ready pair
        acc[half * 4 + jj] = __builtin_amdgcn_wmma_f32_16x16x4_f32(
            false, a, false, b, (short)0, acc[half * 4 + jj], false, false);
      }
    }
  }
  __syncthreads();               // everyone done reading F (pair layout)
  // write T row-major over the LDS buffer (D layout: lanes0-15 M=r, lanes16-31 M=r+8)
#pragma unroll
  for (int half = 0; half < 2; ++half) {
    const int iRow = half ? (7 - p) : p;
#pragma unroll
    for (int jj = 0; jj < 4; ++jj) {
      const int n = (jBase + jj) * 16 + ml;
      v8f d = acc[half * 4 + jj];
#pragma unroll
      for (int r = 0; r < 8; ++r)
        smem[(iRow * 16 + r + dM) * LDS_STRIDE + n] = d[r];
    }
  }
  __syncthreads();

  // ---------- Stage 2: I = T * U   (inclusive row prefix over cols) ---------
#pragma unroll
  for (int t = 0; t < 8; ++t) acc[t] = vzero;

#pragma unroll
  for (int half = 0; half < 2; ++half) {
    const int jCol  = half ? (7 - p) : p;     // output tile col
    const int kbMax = 4 * (jCol + 1);         // U[k,n]=0 for k>n
    const int ng    = jCol * 16 + ml;         // global col (B's N)
    for (int kb = 0; kb < kbMax; ++kb) {
      const int k0 = kb * 4 + kh;             // even
      v2f b;                                   // B = U fragment, generated
      b.x = (k0     <= ng) ? 1.0f : 0.0f;
      b.y = (k0 + 1 <= ng) ? 1.0f : 0.0f;
#pragma unroll
      for (int ii = 0; ii < 4; ++ii) {
        const int m = (jBase + ii) * 16 + ml;  // global row (A's M)
        v2f a = *(const v2f*)&smem[m * LDS_STRIDE + k0];  // contiguous b64, conflict-free
        acc[half * 4 + ii] = __builtin_amdgcn_wmma_f32_16x16x4_f32(
            false, a, false, b, (short)0, acc[half * 4 + ii], false, false);
      }
    }
  }
  __syncthreads();               // everyone done reading T
  // write integral image I row-major over LDS
#pragma unroll
  for (int half = 0; half < 2; ++half) {
    const int jCol = half ? (7 - p) : p;
    const int n    = jCol * 16 + ml;
#pragma unroll
    for (int ii = 0; ii < 4; ++ii) {
      v8f d = acc[half * 4 + ii];
#pragma unroll
      for (int r = 0; r < 8; ++r)
        smem[((jBase + ii) * 16 + r + dM) * LDS_STRIDE + n] = d[r];
    }
  }
  __syncthreads();

  // ---------- Stage 3: per-region corner gather + normalize -----------------
  auto ival = [&](int y, int x) -> float {
    return (y >= 0 && x >= 0) ? smem[y * LDS_STRIDE + x] : 0.0f;
  };
  const float4* roi4 = (const float4*)roi;
#pragma unroll
  for (int q = 0; q < 4; ++q) {
    int n  = q * 256 + tid;        // region index: n = patch*256 + roi
    int ip = n >> 8;               // patch id 0..3 (p=2 grid)
    int r  = n & 255;
    float4 rb = roi4[r];           // xmin, ymin, xmax, ymax
    float ix = (float)(ip & 1);
    float iy = (float)(ip >> 1);
    float ws = (rb.z - rb.x) * 0.5f;
    float hs = (rb.w - rb.y) * 0.5f;
    int x0 = (int)rintf(rb.x + ix * ws);        // rintf == jnp.round (RNE)
    int y0 = (int)rintf(rb.y + iy * hs);
    int x1 = (int)rintf(rb.x + ix * ws + ws);
    int y1 = (int)rintf(rb.y + iy * hs + hs);
    x0 = max(0, min(x0, HW)); x1 = max(0, min(x1, HW));
    y0 = max(0, min(y0, HW)); y1 = max(0, min(y1, HW));
    float s = ival(y1 - 1, x1 - 1) - ival(y0 - 1, x1 - 1)
            - ival(y1 - 1, x0 - 1) + ival(y0 - 1, x0 - 1);
    float cnt = (float)((y1 - y0) * (x1 - x0));
    out[(size_t)n * C_DIM + c] = s / fmaxf(cnt, 1.0f);
  }
}

extern "C" void kernel_launch(void* const* d_in, const int* in_sizes, int n_in,
                              void* d_out, int out_size, void* d_ws, size_t ws_size,
                              hipStream_t stream) {
  const float* fmap = (const float*)d_in[0];   // (1024,128,128) f32
  const float* roi  = (const float*)d_in[1];   // (256,4) f32
  float* out        = (float*)d_out;           // (1024,1024,1,1) f32
  (void)in_sizes; (void)n_in; (void)out_size; (void)d_ws; (void)ws_size;
  roi_pool_integral_wmma<<<C_DIM, 256, 0, stream>>>(fmap, roi, out);
}